// GraphTransformerLayer_43241730736591
// MI455X (gfx1250) — compile-verified
//
#include <hip/hip_runtime.h>

#define DEV __device__ __forceinline__

typedef float v2f __attribute__((ext_vector_type(2)));
typedef float v8f __attribute__((ext_vector_type(8)));

static constexpr int CDIM  = 128;           // model dim
static constexpr int FDIM  = 256;           // ffn dim
static constexpr int NSEQ  = 2048;
static constexpr int BATCH = 4;
static constexpr int NHEAD = 8;
static constexpr int HDIM  = 16;
static constexpr int MROWS = BATCH * NSEQ;  // 8192

DEV v8f v8zero() {
  v8f v;
#pragma unroll
  for (int i = 0; i < 8; ++i) v[i] = 0.f;
  return v;
}

// D = A(16x4,f32) * B(4x16,f32) + C   -> v_wmma_f32_16x16x4_f32
DEV v8f wmma4(v2f a, v2f b, v8f c) {
  return __builtin_amdgcn_wmma_f32_16x16x4_f32(false, a, false, b, (short)0, c,
                                               false, false);
}

// reductions across the 16 lanes of each half-wave (C/D row groups)
DEV float redmax16(float v) {
#pragma unroll
  for (int m = 1; m < 16; m <<= 1) v = fmaxf(v, __shfl_xor(v, m, 32));
  return v;
}
DEV float redsum16(float v) {
#pragma unroll
  for (int m = 1; m < 16; m <<= 1) v += __shfl_xor(v, m, 32);
  return v;
}

// acc[j] (16 x 16 tiles, j = 0..NT-1) += A_strip(16 x K) * B(K x 16*NT)
// Arow: &A[(row0 + (lane&15)) * lda]   Bc: &B[col0 + (lane&15)]  (row-major B)
template <int NT>
DEV void gemm_acc(const float* __restrict__ Arow, const float* __restrict__ Bc,
                  int ldb, int K, v8f* acc) {
  const int hi = (threadIdx.x >> 4) & 1;
  for (int k = 0; k < K; k += 4) {
    const int kb = k + 2 * hi;
    v2f a;
    a.x = Arow[kb];
    a.y = Arow[kb + 1];
#pragma unroll
    for (int j = 0; j < NT; ++j) {
      v2f b;
      b.x = Bc[(size_t)kb * ldb + j * 16];
      b.y = Bc[(size_t)(kb + 1) * ldb + j * 16];
      acc[j] = wmma4(a, b, acc[j]);
    }
  }
}

// ---------------- Kernel 1: Q/K/V projections --------------------------------
__global__ __launch_bounds__(256) void qkv_kernel(
    const float* __restrict__ h, const float* __restrict__ Wq,
    const float* __restrict__ Wk, const float* __restrict__ Wv,
    float* __restrict__ Q, float* __restrict__ K, float* __restrict__ V) {
  const int lane = threadIdx.x & 31;
  const int col = lane & 15, hi = lane >> 4;
  const int gw = blockIdx.x * (blockDim.x >> 5) + (threadIdx.x >> 5);
  const int mat = gw / (MROWS / 16);
  const int rt = gw % (MROWS / 16);
  const float* W = (mat == 0) ? Wq : (mat == 1) ? Wk : Wv;
  float* Out = (mat == 0) ? Q : (mat == 1) ? K : V;
  const int row0 = rt * 16;

  v8f acc[8];
#pragma unroll
  for (int j = 0; j < 8; ++j) acc[j] = v8zero();

  gemm_acc<8>(h + (size_t)(row0 + col) * CDIM, W + col, CDIM, CDIM, acc);

#pragma unroll
  for (int j = 0; j < 8; ++j)
#pragma unroll
    for (int r = 0; r < 8; ++r)
      Out[(size_t)(row0 + r + 8 * hi) * CDIM + 16 * j + col] = acc[j][r];
}

// ---------------- Kernel 2: masked flash attention ---------------------------
// one wave handles one (batch, head, 16-row tile); streams over all 2048 keys
__global__ __launch_bounds__(128) void attn_kernel(
    const float* __restrict__ Q, const float* __restrict__ Kmat,
    const float* __restrict__ V, const int* __restrict__ adj,
    float* __restrict__ O) {
  __shared__ float lds[4][16 * 16];
  const int lane = threadIdx.x & 31;
  const int wid = threadIdx.x >> 5;
  const int col = lane & 15, hi = lane >> 4;
  const int gw = blockIdx.x * 4 + wid;
  const int nt = gw & (NSEQ / 16 - 1);
  const int hd = (gw >> 7) & (NHEAD - 1);
  const int b = gw >> 10;
  const int n0 = nt * 16;
  const size_t base = (size_t)b * NSEQ * CDIM;

  const float* Qrow = Q + base + (size_t)(n0 + col) * CDIM + hd * HDIM;
  float* P = lds[wid];

  float m_i[8], l_i[8];
  v8f o = v8zero();
#pragma unroll
  for (int r = 0; r < 8; ++r) {
    m_i[r] = -__builtin_inff();
    l_i[r] = 0.f;
  }

  for (int m0 = 0; m0 < NSEQ; m0 += 16) {
    // ---- S = Q * K^T  (K of head dim = 16 -> 4 wmma steps) ----
    v8f s = v8zero();
    const float* Kc = Kmat + base + (size_t)(m0 + col) * CDIM + hd * HDIM;
#pragma unroll
    for (int t = 0; t < 4; ++t) {
      const int kb = 4 * t + 2 * hi;
      v2f a, bb;
      a.x = Qrow[kb];
      a.y = Qrow[kb + 1];
      bb.x = Kc[kb];   // B[kk][mcol] = K[m0+mcol][kk] : contiguous pair
      bb.y = Kc[kb + 1];
      s = wmma4(a, bb, s);
    }

    // ---- mask + online softmax (per C-layout row r+8*hi) ----
    const int* adjp = adj + (size_t)(n0 + 8 * hi) * NSEQ + m0 + col;
#pragma unroll
    for (int r = 0; r < 8; ++r) {
      float sv = s[r] * 0.25f;  // 1/sqrt(16)
      if (adjp[(size_t)r * NSEQ] == 0) sv = -1e30f;  // finite -inf sentinel
      const float rmax = redmax16(sv);
      const float mnew = fmaxf(m_i[r], rmax);
      const float f = __expf(m_i[r] - mnew);
      const float p = __expf(sv - mnew);
      l_i[r] = l_i[r] * f + redsum16(p);
      o[r] *= f;
      m_i[r] = mnew;
      P[(r + 8 * hi) * 16 + col] = p;  // re-layout C-format -> row-major LDS
    }
    __builtin_amdgcn_wave_barrier();

    // ---- O += P * V  (P as A-frags from LDS, V as B-frags) ----
    const float* Vc = V + base + (size_t)m0 * CDIM + hd * HDIM + col;
#pragma unroll
    for (int t = 0; t < 4; ++t) {
      const int kb = 4 * t + 2 * hi;
      v2f a, bb;
      a.x = P[col * 16 + kb];  // A row = lane&15
      a.y = P[col * 16 + kb + 1];
      bb.x = Vc[(size_t)kb * CDIM];
      bb.y = Vc[(size_t)(kb + 1) * CDIM];
      o = wmma4(a, bb, o);
    }
    __builtin_amdgcn_wave_barrier();
  }

  float* Op = O + base + (size_t)(n0 + 8 * hi) * CDIM + hd * HDIM + col;
#pragma unroll
  for (int r = 0; r < 8; ++r) Op[(size_t)r * CDIM] = o[r] / l_i[r];
}

// ------ Kernel 3/5: Out = LayerNorm(res + A*W + bias), W is [K,128] ----------
__global__ __launch_bounds__(256) void gemm_ln_kernel(
    const float* __restrict__ A, int lda, int K, const float* __restrict__ W,
    const float* __restrict__ bias, const float* __restrict__ res,
    const float* __restrict__ g, const float* __restrict__ bt,
    float* __restrict__ Out) {
  const int lane = threadIdx.x & 31;
  const int col = lane & 15, hi = lane >> 4;
  const int gw = blockIdx.x * (blockDim.x >> 5) + (threadIdx.x >> 5);
  const int row0 = gw * 16;

  v8f acc[8];
#pragma unroll
  for (int j = 0; j < 8; ++j) acc[j] = v8zero();

  gemm_acc<8>(A + (size_t)(row0 + col) * lda, W + col, CDIM, K, acc);

#pragma unroll
  for (int r = 0; r < 8; ++r) {
    const int row = row0 + r + 8 * hi;
    float s1 = 0.f, s2 = 0.f;
#pragma unroll
    for (int j = 0; j < 8; ++j) {
      float v = acc[j][r] + bias[16 * j + col] +
                res[(size_t)row * CDIM + 16 * j + col];
      acc[j][r] = v;
      s1 += v;
      s2 += v * v;
    }
    s1 = redsum16(s1);
    s2 = redsum16(s2);
    const float mean = s1 * (1.f / CDIM);
    const float var = s2 * (1.f / CDIM) - mean * mean;
    const float rstd = rsqrtf(var + 1e-5f);
#pragma unroll
    for (int j = 0; j < 8; ++j)
      Out[(size_t)row * CDIM + 16 * j + col] =
          (acc[j][r] - mean) * rstd * g[16 * j + col] + bt[16 * j + col];
  }
}

// ---------------- Kernel 4: T = relu(X*W1 + b1), W1 is [128,256] -------------
__global__ __launch_bounds__(256) void ffn1_kernel(
    const float* __restrict__ X, const float* __restrict__ W1,
    const float* __restrict__ b1, float* __restrict__ T) {
  const int lane = threadIdx.x & 31;
  const int col = lane & 15, hi = lane >> 4;
  const int gw = blockIdx.x * (blockDim.x >> 5) + (threadIdx.x >> 5);
  const int half = gw & 1;
  const int rt = gw >> 1;
  const int row0 = rt * 16, c0 = half * 128;

  v8f acc[8];
#pragma unroll
  for (int j = 0; j < 8; ++j) acc[j] = v8zero();

  gemm_acc<8>(X + (size_t)(row0 + col) * CDIM, W1 + c0 + col, FDIM, CDIM, acc);

#pragma unroll
  for (int j = 0; j < 8; ++j)
#pragma unroll
    for (int r = 0; r < 8; ++r)
      T[(size_t)(row0 + r + 8 * hi) * FDIM + c0 + 16 * j + col] =
          fmaxf(acc[j][r] + b1[c0 + 16 * j + col], 0.f);
}

// -----------------------------------------------------------------------------
extern "C" void kernel_launch(void* const* d_in, const int* in_sizes, int n_in,
                              void* d_out, int out_size, void* d_ws,
                              size_t ws_size, hipStream_t stream) {
  (void)in_sizes; (void)n_in; (void)out_size; (void)ws_size;
  const int* adj  = (const int*)d_in[0];
  const float* h  = (const float*)d_in[1];
  const float* Wq = (const float*)d_in[2];
  const float* Wk = (const float*)d_in[3];
  const float* Wv = (const float*)d_in[4];
  const float* Wo = (const float*)d_in[5];
  const float* bo = (const float*)d_in[6];
  const float* g1 = (const float*)d_in[7];
  const float* bt1 = (const float*)d_in[8];
  const float* W1 = (const float*)d_in[9];
  const float* bb1 = (const float*)d_in[10];
  const float* W2 = (const float*)d_in[11];
  const float* bb2 = (const float*)d_in[12];
  const float* g2 = (const float*)d_in[13];
  const float* bt2 = (const float*)d_in[14];

  float* ws = (float*)d_ws;
  const size_t SZ = (size_t)MROWS * CDIM;  // 1M floats
  float* Q  = ws;
  float* K  = Q + SZ;
  float* V  = K + SZ;
  float* AO = V + SZ;
  float* X  = AO + SZ;
  float* T  = X + SZ;  // 2M floats  (total ~28 MB)

  // 1) QKV: 1536 waves (512 row-tiles x 3 matrices), 8 waves/block
  qkv_kernel<<<(3 * (MROWS / 16)) / 8, 256, 0, stream>>>(h, Wq, Wk, Wv, Q, K, V);
  // 2) attention: 4096 waves (4 x 8 x 128 tiles), 4 waves/block
  attn_kernel<<<(BATCH * NHEAD * (NSEQ / 16)) / 4, 128, 0, stream>>>(Q, K, V,
                                                                     adj, AO);
  // 3) x = LN1(h + AO*Wo + bo)
  gemm_ln_kernel<<<(MROWS / 16) / 8, 256, 0, stream>>>(AO, CDIM, CDIM, Wo, bo,
                                                       h, g1, bt1, X);
  // 4) t = relu(x*W1 + b1)
  ffn1_kernel<<<(2 * (MROWS / 16)) / 8, 256, 0, stream>>>(X, W1, bb1, T);
  // 5) out = LN2(x + t*W2 + b2)
  gemm_ln_kernel<<<(MROWS / 16) / 8, 256, 0, stream>>>(T, FDIM, FDIM, W2, bb2,
                                                       X, g2, bt2,
                                                       (float*)d_out);
}